// SpatialMemoryGrid_85830626443560
// MI455X (gfx1250) — compile-verified
//
#include <hip/hip_runtime.h>
#include <stdint.h>

// Problem constants (match reference)
#define B_     4
#define GH_    64
#define GW_    64
#define NOBJ_  16
#define FEAT_  256
#define OUTF_  258                       // FEAT + conf + temporal
#define NCELLS_ (B_*GH_*GW_*NOBJ_)       // 262144 cells

// Bulk-copy tiling
#define CPB    32                        // cells per block  (32 KB LDS of features)
#define TPB    256                       // threads per block (8 waves, wave32)
#define CHUNKS (CPB*FEAT_/4)             // 2048 x 16B chunks per block
#define CPT    (CHUNKS/TPB)              // 8 chunks per thread

// ---- CDNA5 async global<->LDS path (guarded; falls back to plain copy) ----
#if defined(__HIP_DEVICE_COMPILE__) && \
    __has_builtin(__builtin_amdgcn_global_load_async_to_lds_b128) && \
    __has_builtin(__builtin_amdgcn_global_store_async_from_lds_b64)
#define USE_ASYNC 1
#else
#define USE_ASYNC 0
#endif

#if USE_ASYNC
#define AS1 __attribute__((address_space(1)))
#define AS3 __attribute__((address_space(3)))

// Match clang builtin prototypes exactly: V4i / V2i are GCC-style vector types.
typedef int v4i __attribute__((vector_size(16)));
typedef int v2i __attribute__((vector_size(8)));

__device__ __forceinline__ AS1 v4i* g4(const void* p) {
    return (AS1 v4i*)(unsigned long long)(uintptr_t)p;        // inttoptr, AS1
}
__device__ __forceinline__ AS1 v2i* g2(const void* p) {
    return (AS1 v2i*)(unsigned long long)(uintptr_t)p;
}
__device__ __forceinline__ AS3 v4i* l4(const void* p) {
    // low 32 bits of a generic LDS pointer are the LDS byte offset
    return (AS3 v4i*)(unsigned int)(uintptr_t)p;
}
__device__ __forceinline__ AS3 v2i* l2(const void* p) {
    return (AS3 v2i*)(unsigned int)(uintptr_t)p;
}
__device__ __forceinline__ void wait_async0() {
#if __has_builtin(__builtin_amdgcn_s_wait_asynccnt)
    __builtin_amdgcn_s_wait_asynccnt(0);
#else
    asm volatile("s_wait_asynccnt 0" ::: "memory");
#endif
}
#endif

// -----------------------------------------------------------------------------
// Kernel 1: bulk repack. out[cell, 0:256] = grid_state[cell, :]
//           out[cell, 256]   = conf[cell] * 0.95
//           out[cell, 257]   = temporal[cell]
// Each block handles CPB consecutive cells; features staged global->LDS->global
// via CDNA5 async copy instructions (ASYNCcnt), no VGPR data movement.
// -----------------------------------------------------------------------------
__global__ __launch_bounds__(TPB)
void smg_bulk_kernel(const float* __restrict__ gs,
                     const float* __restrict__ conf,
                     const float* __restrict__ temp,
                     float* __restrict__ out)
{
    __shared__ __align__(16) float smem[CPB * FEAT_];

    const int       t     = threadIdx.x;
    const long long cell0 = (long long)blockIdx.x * CPB;
    const float*    src   = gs + cell0 * FEAT_;

#if USE_ASYNC
    // Phase 1: async-load this block's 32 KB of features into LDS (B128 chunks).
    #pragma unroll
    for (int j = 0; j < CPT; ++j) {
        const int c = t + j * TPB;                 // 16B chunk id within block
        __builtin_amdgcn_global_load_async_to_lds_b128(
            g4(src + (size_t)c * 4), l4(&smem[c * 4]), 0, 0);
    }
    // Each thread only stores chunks it loaded itself -> per-wave wait suffices.
    wait_async0();

    // Phase 2: async-store into the 258-stride output (8B aligned -> B64).
    #pragma unroll
    for (int j = 0; j < CPT; ++j) {
        const int c  = t + j * TPB;
        const int cl = c >> 6;                     // local cell = c / (FEAT/4)
        const int k  = (c & 63) * 4;               // feature offset within cell
        float* dst = out + (cell0 + cl) * OUTF_ + k;
        __builtin_amdgcn_global_store_async_from_lds_b64(
            g2(dst),     l2(&smem[c * 4]),     0, 0);
        __builtin_amdgcn_global_store_async_from_lds_b64(
            g2(dst + 2), l2(&smem[c * 4 + 2]), 0, 0);
    }
#else
    // Fallback: direct VGPR copy, B128 loads + 2x B64 stores.
    #pragma unroll
    for (int j = 0; j < CPT; ++j) {
        const int c  = t + j * TPB;
        const int cl = c >> 6;
        const int k  = (c & 63) * 4;
        const float4 v = *(const float4*)(src + (size_t)c * 4);
        float* dst = out + (cell0 + cl) * OUTF_ + k;
        float2 lo; lo.x = v.x; lo.y = v.y;
        float2 hi; hi.x = v.z; hi.y = v.w;
        *(float2*)(dst)     = lo;
        *(float2*)(dst + 2) = hi;
    }
    (void)smem;
#endif

    // conf/temporal tail: one float2 per cell (8B aligned: cell*1032 + 1024).
    if (t < CPB) {
        const long long cell = cell0 + t;
        float2 ct;
        ct.x = conf[cell] * 0.95f;     // global decay
        ct.y = temp[cell];
        *(float2*)(out + cell * OUTF_ + FEAT_) = ct;
    }

#if USE_ASYNC
    wait_async0();                      // drain async stores before ENDPGM
#endif
}

// -----------------------------------------------------------------------------
// Kernel 2: overwrite the B*N = 64 scattered target cells (runs after bulk).
// One block per (b, obj), 64 threads; indices unique per (b, obj) -> no races.
// -----------------------------------------------------------------------------
__global__ __launch_bounds__(64)
void smg_fixup_kernel(const float* __restrict__ gs,
                      const float* __restrict__ conf,
                      const float* __restrict__ temp,
                      const float* __restrict__ objf,
                      const float* __restrict__ pos,
                      const float* __restrict__ occ,
                      float* __restrict__ out)
{
    const int m = blockIdx.x;          // 0..63
    const int b = m >> 4;
    const int o = m & (NOBJ_ - 1);
    const int t = threadIdx.x;         // 0..63

    const float x = pos[(b * NOBJ_ + o) * 2 + 0];
    const float y = pos[(b * NOBJ_ + o) * 2 + 1];
    // position_to_grid: scale by (dim-1), clamp as float, truncate to int
    const int gwi = (int)fminf(fmaxf(x * (float)(GW_ - 1), 0.0f), (float)(GW_ - 1));
    const int ghi = (int)fminf(fmaxf(y * (float)(GH_ - 1), 0.0f), (float)(GH_ - 1));
    const long long cell = (((long long)b * GH_ + ghi) * GW_ + gwi) * NOBJ_ + o;

    const bool  vis   = occ[b * NOBJ_ + o] < 0.5f;
    const float alpha = vis ? 0.8f : 0.3f;
    const float beta  = 1.0f - alpha;

    const float* cf  = gs   + cell * FEAT_;
    const float* of  = objf + (size_t)(b * NOBJ_ + o) * FEAT_;
    float*       dst = out  + cell * OUTF_;

    #pragma unroll
    for (int j = 0; j < 2; ++j) {
        const int k = (t + j * 64) * 2;               // 128 float2 per cell
        const float2 cur = *(const float2*)(cf + k);
        const float2 ob  = *(const float2*)(of + k);
        float2 nf;
        nf.x = fmaf(alpha, ob.x, beta * cur.x);
        nf.y = fmaf(alpha, ob.y, beta * cur.y);
        *(float2*)(dst + k) = nf;
    }

    if (t == 0) {
        const float cc = conf[cell];
        const float nc = vis ? fminf(1.0f, fmaf(cc, 0.9f, 0.5f)) : cc * 0.95f;
        float2 ct;
        ct.x = nc * 0.95f;                             // .set(new_conf) then *DECAY
        ct.y = temp[cell] + (vis ? 1.0f : 0.5f);       // .add(temporal_inc)
        *(float2*)(dst + FEAT_) = ct;
    }
}

// -----------------------------------------------------------------------------
extern "C" void kernel_launch(void* const* d_in, const int* in_sizes, int n_in,
                              void* d_out, int out_size, void* d_ws, size_t ws_size,
                              hipStream_t stream) {
    const float* gs   = (const float*)d_in[0];  // grid_state      [B,GH,GW,N,F]
    const float* conf = (const float*)d_in[1];  // grid_confidence [B,GH,GW,N]
    const float* temp = (const float*)d_in[2];  // grid_temporal   [B,GH,GW,N]
    const float* objf = (const float*)d_in[3];  // object_features [B,N,F]
    const float* pos  = (const float*)d_in[4];  // positions       [B,N,2]
    const float* occ  = (const float*)d_in[5];  // occlusion       [B,N]
    float* out = (float*)d_out;                 // [B,GH,GW,N,F+2]

    smg_bulk_kernel<<<NCELLS_ / CPB, TPB, 0, stream>>>(gs, conf, temp, out);
    smg_fixup_kernel<<<B_ * NOBJ_, 64, 0, stream>>>(gs, conf, temp, objf, pos, occ, out);
}